// CoralLoss_1082331758748
// MI455X (gfx1250) — compile-verified
//
#include <hip/hip_runtime.h>
#include <hip/hip_bf16.h>
#include <math.h>

// ---------------------------------------------------------------------------
// Stablemax-CE + halting loss for logits [4,1024,32000] f32.
// Memory-bound streaming reduction: 524 MB read once -> ~22.5us floor at
// 23.3 TB/s. No matrix ops in this workload; the CDNA5 path that matters is
// the async global->LDS engine (ASYNCcnt): 8-deep ring of 4 KB chunks per
// block keeps ~32 KB in flight per block (~256 KB/WGP with 8 resident
// blocks), covering the HBM latency-bandwidth product. Loads are issued
// non-temporal (read-once, 524 MB >> 192 MB L2).
// ---------------------------------------------------------------------------

#define B_    4
#define L_    1024
#define V_    32000
#define TOK_  (B_ * L_)
#define TPB_  256
#define CHUNK_FLOATS (TPB_ * 4)                         // 1024 floats = 4 KB
#define NCHUNK ((V_ + CHUNK_FLOATS - 1) / CHUNK_FLOATS) // 32 (last partial)
#define DEPTH 8                                         // async pipeline depth
#define IGNORE_LABEL (-100LL)
#define CPOL_NT 1                                       // TH_LOAD_NT

typedef int v4i __attribute__((ext_vector_type(4)));
typedef __attribute__((address_space(1))) v4i as1_v4i;   // global AS
typedef __attribute__((address_space(3))) v4i as3_v4i;   // LDS AS

// ---- async global -> LDS b128 copy (one instruction per wave, 512B) -------
__device__ __forceinline__ void async_copy_b128(const float* gsrc, void* ldst) {
#if defined(__has_builtin) && __has_builtin(__builtin_amdgcn_global_load_async_to_lds_b128)
  __builtin_amdgcn_global_load_async_to_lds_b128(
      (as1_v4i*)gsrc, (as3_v4i*)ldst, /*imm offset*/0, /*cpol*/CPOL_NT);
#else
  unsigned lds_off = (unsigned)(unsigned long long)(as3_v4i*)ldst;
  unsigned long long ga = (unsigned long long)gsrc;
  asm volatile("global_load_async_to_lds_b128 %0, %1, off th:TH_LOAD_NT"
               :: "v"(lds_off), "v"(ga) : "memory");
#endif
}

__device__ __forceinline__ void wait_async_le7() {
#if defined(__has_builtin) && __has_builtin(__builtin_amdgcn_s_wait_asynccnt)
  __builtin_amdgcn_s_wait_asynccnt(DEPTH - 1);
#else
  asm volatile("s_wait_asynccnt 7" ::: "memory");
#endif
  asm volatile("" ::: "memory");
}

__device__ __forceinline__ void wait_async_le0() {
#if defined(__has_builtin) && __has_builtin(__builtin_amdgcn_s_wait_asynccnt)
  __builtin_amdgcn_s_wait_asynccnt(0);
#else
  asm volatile("s_wait_asynccnt 0" ::: "memory");
#endif
  asm volatile("" ::: "memory");
}

// ---------------------------------------------------------------------------
// Kernel 1: one block per token. Streams 125 KB row through an 8-deep LDS
// ring via async copies; computes sum(s), argmax, s[label].
// ---------------------------------------------------------------------------
__global__ __launch_bounds__(TPB_)
void coral_token_reduce(const float* __restrict__ logits,
                        const long long* __restrict__ labels,
                        double* __restrict__ tok_loss,
                        int* __restrict__ tok_correct,
                        int* __restrict__ tok_valid) {
  __shared__ float  stage[DEPTH][CHUNK_FLOATS];   // 32 KB staging ring
  __shared__ double red_sum[TPB_];
  __shared__ float  red_max[TPB_];
  __shared__ int    red_idx[TPB_];
  __shared__ double s_label_sh;

  const int tok = blockIdx.x;
  const int tid = threadIdx.x;
  const long long lab = labels[tok];
  const float* row = logits + (size_t)tok * V_;

  if (tid == 0) s_label_sh = 0.0;
  __syncthreads();

  // Prologue: fill pipeline. Lane-private LDS slot = stage[buf][tid*4].
  #pragma unroll
  for (int c = 0; c < DEPTH; ++c) {
    int e0 = c * CHUNK_FLOATS + tid * 4;
    int ce = e0 > (V_ - 4) ? (V_ - 4) : e0;   // clamp for the partial tail
    async_copy_b128(row + ce, &stage[c][tid * 4]);
  }

  double sumd = 0.0;
  float  maxv = -__builtin_inff();
  int    maxi = 0x7fffffff;
  double slab = 0.0;

  for (int c = 0; c < NCHUNK; ++c) {
    wait_async_le7();                            // oldest chunk now in LDS
    float4 v = *(const float4*)&stage[c % DEPTH][tid * 4];
    const int e0 = c * CHUNK_FLOATS + tid * 4;
    float xs[4] = {v.x, v.y, v.z, v.w};
    float gs = 0.f;
    #pragma unroll
    for (int j = 0; j < 4; ++j) {
      int e = e0 + j;
      if (e < V_) {
        float x = xs[j];
        float s = (x < 0.f) ? (1.f / (1.f - x + 1e-30f)) : (x + 1.f);
        gs += s;
        if (x > maxv) { maxv = x; maxi = e; }    // per-thread first occurrence
        if ((long long)e == lab) slab = (double)s;
      }
    }
    sumd += (double)gs;                          // f64 accumulation as in ref
    int cn = c + DEPTH;
    if (cn < NCHUNK) {
      int e0n = cn * CHUNK_FLOATS + tid * 4;
      int ce  = e0n > (V_ - 4) ? (V_ - 4) : e0n;
      async_copy_b128(row + ce, &stage[cn % DEPTH][tid * 4]);
    }
  }
  wait_async_le0();

  if (slab != 0.0) s_label_sh = slab;            // exactly one thread matches
  red_sum[tid] = sumd;
  red_max[tid] = maxv;
  red_idx[tid] = maxi;
  __syncthreads();

  for (int s = TPB_ / 2; s > 0; s >>= 1) {
    if (tid < s) {
      red_sum[tid] += red_sum[tid + s];
      float mo = red_max[tid + s]; int io = red_idx[tid + s];
      if (mo > red_max[tid] || (mo == red_max[tid] && io < red_idx[tid])) {
        red_max[tid] = mo; red_idx[tid] = io;    // argmax: first index on tie
      }
    }
    __syncthreads();
  }

  if (tid == 0) {
    int valid = (lab != IGNORE_LABEL) ? 1 : 0;
    double loss = 0.0;
    if (valid) loss = log(red_sum[0]) - log(s_label_sh);
    tok_loss[tok]    = loss;
    tok_valid[tok]   = valid;
    tok_correct[tok] = (valid && ((long long)red_idx[0] == lab)) ? 1 : 0;
  }
}

// ---------------------------------------------------------------------------
// Kernel 2: per-batch reduce (64 threads/batch) + halting losses -> scalar.
// ---------------------------------------------------------------------------
__global__ __launch_bounds__(256)
void coral_finalize(const double* __restrict__ tok_loss,
                    const int* __restrict__ tok_correct,
                    const int* __restrict__ tok_valid,
                    const float* __restrict__ qh,
                    const float* __restrict__ qc,
                    float* __restrict__ out) {
  __shared__ double lsum[256];
  __shared__ int    vsum[256];
  __shared__ int    csum[256];
  const int tid  = threadIdx.x;
  const int b    = tid >> 6;    // 4 batches x 64 threads
  const int lane = tid & 63;

  double ls = 0.0; int vs = 0, cs = 0;
  for (int i = lane; i < L_; i += 64) {
    int t = b * L_ + i;
    ls += tok_loss[t]; vs += tok_valid[t]; cs += tok_correct[t];
  }
  lsum[tid] = ls; vsum[tid] = vs; csum[tid] = cs;
  __syncthreads();
  for (int s = 32; s > 0; s >>= 1) {
    if (lane < s) {
      lsum[tid] += lsum[tid + s];
      vsum[tid] += vsum[tid + s];
      csum[tid] += csum[tid + s];
    }
    __syncthreads();
  }

  if (tid == 0) {
    float ltask = 0.f, bce_h = 0.f, bce_c = 0.f;
    for (int bb = 0; bb < B_; ++bb) {
      double sl = lsum[bb * 64];
      int vv = vsum[bb * 64], cc = csum[bb * 64];
      float cnt = (float)(vv > 1 ? vv : 1);
      ltask += (float)sl / cnt;
      float tgt = (cc == vv) ? 1.f : 0.f;                 // seq_is_correct
      float x = qh[bb];
      bce_h += fmaxf(x, 0.f) - x * tgt + log1pf(expf(-fabsf(x)));
      float tc = 1.f / (1.f + expf(-x));                  // sigmoid(q_halt)
      float y = qc[bb];
      bce_c += fmaxf(y, 0.f) - y * tc + log1pf(expf(-fabsf(y)));
    }
    ltask *= (1.f / B_); bce_h *= (1.f / B_); bce_c *= (1.f / B_);
    out[0] = ltask + 0.5f * (bce_h + bce_c);
  }
}

// ---------------------------------------------------------------------------
extern "C" void kernel_launch(void* const* d_in, const int* in_sizes, int n_in,
                              void* d_out, int out_size, void* d_ws, size_t ws_size,
                              hipStream_t stream) {
  const float* logits     = (const float*)d_in[0];
  const float* q_halt     = (const float*)d_in[1];
  const float* q_continue = (const float*)d_in[2];
  const long long* labels = (const long long*)d_in[3]; // jnp int64 (x64 mode)

  double* tok_loss   = (double*)d_ws;                               // 32 KB
  int*    tok_corr   = (int*)((char*)d_ws + TOK_ * sizeof(double)); // 16 KB
  int*    tok_valid  = tok_corr + TOK_;                             // 16 KB

  coral_token_reduce<<<TOK_, TPB_, 0, stream>>>(logits, labels,
                                                tok_loss, tok_corr, tok_valid);
  coral_finalize<<<1, 256, 0, stream>>>(tok_loss, tok_corr, tok_valid,
                                        q_halt, q_continue, (float*)d_out);
}